// GeneratorLSTM_3736621548192
// MI455X (gfx1250) — compile-verified
//
#include <hip/hip_runtime.h>
#include <math.h>

// ---------------------------------------------------------------------------
// CDNA5 (gfx1250) implementation.
// Dominant work: two 4096^3 FP32 GEMMs -> native V_WMMA_F32_16X16X4_F32 path.
// Wave32. Workgroup = 256 threads = 8 waves. Block tile 128x64; each wave
// owns a 32x32 tile = 2x2 outer product of 16x16 f32 accumulators, so 4 WMMAs
// issue per A/B fragment pair. LDS double-buffered (one barrier per K-chunk);
// B is stored K-pair-interleaved so every B fragment is a single even-aligned
// ds_load_b64 consumed directly by the WMMA (no repack moves).
// ---------------------------------------------------------------------------

typedef __attribute__((ext_vector_type(2))) float v2f;
typedef __attribute__((ext_vector_type(8))) float v8f;

__device__ __forceinline__ float sigmoidf_(float x) { return 1.0f / (1.0f + __expf(-x)); }

// Orderable key for float argmax via integer atomicMax.
__device__ __forceinline__ unsigned fkey_(float f) {
  unsigned u = __float_as_uint(f);
  return (u & 0x80000000u) ? ~u : (u | 0x80000000u);
}

// ---------------------------------------------------------------------------
// Tiled FP32 WMMA GEMM: C[M,N] = act(A[M,K] @ B[K,N] + bias)
// Requires M%128==0, N%64==0, K%16==0 (true for all WMMA GEMMs here).
// ACT: 0=none, 1=relu, 2=sigmoid.  bias may be nullptr.
// A-fragment (16x4 f32): lanes 0-15 hold K=0/1 in v0/v1, lanes 16-31 K=2/3.
// B-fragment (4x16 f32): v0 = rows K=0 (lanes 0-15) / K=2 (lanes 16-31),
//                        v1 = rows K=1 / K=3.
// ---------------------------------------------------------------------------
template <int ACT>
__global__ __launch_bounds__(256) void wmma_gemm_f32(
    const float* __restrict__ A, const float* __restrict__ B,
    const float* __restrict__ bias, float* __restrict__ C,
    int M, int N, int K) {
  __shared__ float As[2][128][17];    // row-major, pad 17 (conflict-free)
  __shared__ float Bs[2][8][66][2];   // [K-pair][col][2]: fragment = b64 load

  const int tid    = threadIdx.x;
  const int wave   = tid >> 5;
  const int lane   = tid & 31;
  const int lane16 = lane & 15;
  const int lhi    = lane >> 4;        // 0: K=0/1 rows, 1: K=2/3 rows
  const int wm     = wave & 3;         // 4 row-tiles of 32
  const int wn     = wave >> 2;        // 2 col-tiles of 32
  const int rowBase = blockIdx.y * 128;
  const int colBase = blockIdx.x * 64;

  // Staging coordinates. A: 2 float4/thread (row ar, cols ac..ac+7).
  // B: thread covers K-rows {2*brp, 2*brp+1} x cols {bc, bc+1} -> one b128.
  const int ar = tid >> 1, ac = (tid & 1) << 3;  // A chunk 128x16
  const int brp = tid >> 5, bc = (tid & 31) << 1; // B chunk 16x64 as 8 pairs

  v8f acc00 = {}, acc01 = {}, acc10 = {}, acc11 = {};

  // Prologue: stage K-chunk 0.
  {
    const float4 a0 = *(const float4*)(A + (size_t)(rowBase + ar) * K + ac);
    const float4 a1 = *(const float4*)(A + (size_t)(rowBase + ar) * K + ac + 4);
    const float2 g0 = *(const float2*)(B + (size_t)(2 * brp)     * N + colBase + bc);
    const float2 g1 = *(const float2*)(B + (size_t)(2 * brp + 1) * N + colBase + bc);
    As[0][ar][ac + 0] = a0.x; As[0][ar][ac + 1] = a0.y;
    As[0][ar][ac + 2] = a0.z; As[0][ar][ac + 3] = a0.w;
    As[0][ar][ac + 4] = a1.x; As[0][ar][ac + 5] = a1.y;
    As[0][ar][ac + 6] = a1.z; As[0][ar][ac + 7] = a1.w;
    float4 bv; bv.x = g0.x; bv.y = g1.x; bv.z = g0.y; bv.w = g1.y;
    *(float4*)&Bs[0][brp][bc][0] = bv;
  }
  __syncthreads();

  const int nc = K >> 4;
  for (int c = 0; c < nc; ++c) {
    const int cur = c & 1;
    const bool more = (c + 1 < nc);
    float4 ra0 = {}, ra1 = {};
    float2 rg0 = {}, rg1 = {};
    if (more) {  // issue next-chunk global loads; latency hides under WMMAs
      const int k0 = (c + 1) << 4;
      ra0 = *(const float4*)(A + (size_t)(rowBase + ar) * K + k0 + ac);
      ra1 = *(const float4*)(A + (size_t)(rowBase + ar) * K + k0 + ac + 4);
      rg0 = *(const float2*)(B + (size_t)(k0 + 2 * brp)     * N + colBase + bc);
      rg1 = *(const float2*)(B + (size_t)(k0 + 2 * brp + 1) * N + colBase + bc);
    }

#pragma unroll
    for (int kk = 0; kk < 16; kk += 4) {
      const int kp = (kk >> 1) + lhi;  // K-pair index for this half-wave
      v2f a0, a1;
      a0.x = As[cur][wm * 32 + lane16][kk + 2 * lhi];
      a0.y = As[cur][wm * 32 + lane16][kk + 2 * lhi + 1];
      a1.x = As[cur][wm * 32 + 16 + lane16][kk + 2 * lhi];
      a1.y = As[cur][wm * 32 + 16 + lane16][kk + 2 * lhi + 1];
      const v2f b0 = *(const v2f*)&Bs[cur][kp][wn * 32 + lane16][0];
      const v2f b1 = *(const v2f*)&Bs[cur][kp][wn * 32 + 16 + lane16][0];
      acc00 = __builtin_amdgcn_wmma_f32_16x16x4_f32(false, a0, false, b0,
                                                    (short)0, acc00, false, false);
      acc01 = __builtin_amdgcn_wmma_f32_16x16x4_f32(false, a0, false, b1,
                                                    (short)0, acc01, false, false);
      acc10 = __builtin_amdgcn_wmma_f32_16x16x4_f32(false, a1, false, b0,
                                                    (short)0, acc10, false, false);
      acc11 = __builtin_amdgcn_wmma_f32_16x16x4_f32(false, a1, false, b1,
                                                    (short)0, acc11, false, false);
    }

    if (more) {  // commit staged chunk to the other LDS buffer
      const int nxt = cur ^ 1;
      As[nxt][ar][ac + 0] = ra0.x; As[nxt][ar][ac + 1] = ra0.y;
      As[nxt][ar][ac + 2] = ra0.z; As[nxt][ar][ac + 3] = ra0.w;
      As[nxt][ar][ac + 4] = ra1.x; As[nxt][ar][ac + 5] = ra1.y;
      As[nxt][ar][ac + 6] = ra1.z; As[nxt][ar][ac + 7] = ra1.w;
      float4 bv; bv.x = rg0.x; bv.y = rg1.x; bv.z = rg0.y; bv.w = rg1.y;
      *(float4*)&Bs[nxt][brp][bc][0] = bv;
      __syncthreads();
    }
  }

  // Epilogue. C/D layout: VGPR r, lanes 0-15 -> M=r, lanes 16-31 -> M=r+8.
  const int c0 = colBase + wn * 32 + lane16;
#pragma unroll
  for (int r = 0; r < 8; ++r) {
    const int row0 = rowBase + wm * 32 + r + 8 * lhi;
    const int row1 = row0 + 16;
    float v00 = acc00[r], v01 = acc01[r], v10 = acc10[r], v11 = acc11[r];
    if (bias) {
      const float bA = bias[c0], bB = bias[c0 + 16];
      v00 += bA; v01 += bB; v10 += bA; v11 += bB;
    }
    if (ACT == 1) {
      v00 = fmaxf(v00, 0.f); v01 = fmaxf(v01, 0.f);
      v10 = fmaxf(v10, 0.f); v11 = fmaxf(v11, 0.f);
    } else if (ACT == 2) {
      v00 = sigmoidf_(v00); v01 = sigmoidf_(v01);
      v10 = sigmoidf_(v10); v11 = sigmoidf_(v11);
    }
    C[(size_t)row0 * N + c0]      = v00;
    C[(size_t)row0 * N + c0 + 16] = v01;
    C[(size_t)row1 * N + c0]      = v10;
    C[(size_t)row1 * N + c0 + 16] = v11;
  }
}

// h_graph0[j] = sum_m gate[m,j] * graph_lin[m,j]   (j < 1024, coalesced over j)
__global__ void gated_colsum_kernel(const float* __restrict__ g,
                                    const float* __restrict__ p,
                                    float* __restrict__ out) {
  const int j = blockIdx.x * blockDim.x + threadIdx.x;
  float s = 0.f;
  for (int m = 0; m < 4096; ++m)
    s += g[(size_t)m * 1024 + j] * p[(size_t)m * 1024 + j];
  out[j] = s;
}

// seq_embed = x(512x20) @ W_seq(20x64) + b_seq
__global__ void seq_embed_kernel(const float* __restrict__ x,
                                 const float* __restrict__ W,
                                 const float* __restrict__ b,
                                 float* __restrict__ out) {
  const int idx = blockIdx.x * blockDim.x + threadIdx.x;  // 512*64
  const int n = idx >> 6, i = idx & 63;
  float s = b[i];
#pragma unroll
  for (int a = 0; a < 20; ++a) s += x[n * 20 + a] * W[a * 64 + i];
  out[idx] = s;
}

// One LSTM step (single workgroup). Also resets this step's argmax slot.
__global__ __launch_bounds__(1024) void lstm_cell_kernel(
    const float* __restrict__ seq_embed, const float* __restrict__ W_ih,
    const float* __restrict__ W_hh, const float* __restrict__ b_ih,
    const float* __restrict__ b_hh, float* __restrict__ hh,
    float* __restrict__ cc, unsigned long long* __restrict__ amax, int t) {
  __shared__ float xs[64];
  __shared__ float hs[512];
  __shared__ float gates[2048];
  const int tid = threadIdx.x;
  if (tid < 64)  xs[tid] = seq_embed[t * 64 + tid];
  if (tid < 512) hs[tid] = hh[tid];
  if (tid == 0)  amax[t] = 0ull;
  __syncthreads();

  for (int j = tid; j < 2048; j += 1024) {
    float acc = b_ih[j] + b_hh[j];
    for (int a = 0; a < 64; ++a)  acc += xs[a] * W_ih[a * 2048 + j];
    for (int g = 0; g < 512; ++g) acc += hs[g] * W_hh[g * 2048 + j];
    gates[j] = acc;
  }
  __syncthreads();

  if (tid < 512) {
    const float gi = gates[tid], gf = gates[512 + tid];
    const float gg = gates[1024 + tid], go = gates[1536 + tid];
    const float c = sigmoidf_(gf) * cc[tid] + sigmoidf_(gi) * tanhf(gg);
    const float h = sigmoidf_(go) * tanhf(c);
    cc[tid] = c;
    hh[tid] = h;
  }
}

// z[m] = relu(node_proj[m,:] + hh) . w_ge + b_ge ; scores + packed argmax.
// One wave per node, lane-parallel dot over G=512, shuffle reduction.
__global__ __launch_bounds__(256) void z_argmax_kernel(
    const float* __restrict__ node_proj, const float* __restrict__ hh,
    const float* __restrict__ w_ge, const float* __restrict__ b_ge,
    float* __restrict__ scores, unsigned long long* __restrict__ amax, int t) {
  __shared__ float hs[512];
  __shared__ float wg[512];
  const int tid = threadIdx.x;
  for (int i = tid; i < 512; i += 256) { hs[i] = hh[i]; wg[i] = w_ge[i]; }
  __syncthreads();

  const int wave = tid >> 5, lane = tid & 31;
  const int m = blockIdx.x * 8 + wave;
  const float* row = node_proj + (size_t)m * 512;
  float s = 0.f;
#pragma unroll
  for (int g = lane; g < 512; g += 32)
    s += fmaxf(row[g] + hs[g], 0.f) * wg[g];
  for (int off = 16; off; off >>= 1) s += __shfl_xor(s, off, 32);

  if (lane == 0) {
    const float z = s + b_ge[0];
    scores[(size_t)t * 4096 + m] = z;
    // Inverted index in low bits: at equal score, smaller m wins (jnp.argmax).
    const unsigned long long packed =
        ((unsigned long long)fkey_(z) << 32) | (unsigned)(4095 - m);
    atomicMax(amax + t, packed);
  }
}

__global__ void idx_finalize_kernel(const unsigned long long* __restrict__ amax,
                                    float* __restrict__ out_idx) {
  const int t = blockIdx.x * blockDim.x + threadIdx.x;
  if (t < 512) {
    const unsigned inv = (unsigned)(amax[t] & 0xFFFFFFFFull);
    out_idx[t] = (float)(4095u - inv);
  }
}

// ---------------------------------------------------------------------------
// Workspace layout (floats). Needs ~134.4 MB.
//   [0,        16M)  t0 = d@a        -> reused after step 2 for activations
//   [16M,      32M)  dmat = t0@d     (live through step 6)
//   [32M,      ...)  tail: h_graph0(1024)=h||c, seq_embed(32768), amax(512 u64)
// ---------------------------------------------------------------------------
extern "C" void kernel_launch(void* const* d_in, const int* in_sizes, int n_in,
                              void* d_out, int out_size, void* d_ws, size_t ws_size,
                              hipStream_t stream) {
  const float* x_tensor = (const float*)d_in[0];
  const float* f_tensor = (const float*)d_in[1];
  const float* a_tensor = (const float*)d_in[2];
  const float* d_tensor = (const float*)d_in[3];
  const float* W_feat   = (const float*)d_in[4];
  const float* b_feat   = (const float*)d_in[5];
  const float* W_gcn1   = (const float*)d_in[6];
  const float* W_gcn2   = (const float*)d_in[7];
  const float* W_ih     = (const float*)d_in[8];
  const float* W_hh     = (const float*)d_in[9];
  const float* b_ih     = (const float*)d_in[10];
  const float* b_hh     = (const float*)d_in[11];
  const float* W_ne     = (const float*)d_in[12];
  const float* b_ne     = (const float*)d_in[13];
  const float* w_ge     = (const float*)d_in[14];
  const float* b_ge     = (const float*)d_in[15];
  const float* W_gate   = (const float*)d_in[16];
  const float* b_gate   = (const float*)d_in[17];
  const float* W_graph  = (const float*)d_in[18];
  const float* b_graph  = (const float*)d_in[19];
  const float* W_seq    = (const float*)d_in[20];
  const float* b_seq    = (const float*)d_in[21];

  float* ws = (float*)d_ws;
  const size_t MM = 4096ull * 4096ull;      // 16M floats
  float* t0   = ws;
  float* dmat = ws + MM;
  // Region-A reuse (t0 is dead after dmat is produced):
  float* f_nodes = ws + 0ull  * 1048576;    // 4096x256
  float* t1      = ws + 1ull  * 1048576;    // 4096x256
  float* relu_h  = ws + 2ull  * 1048576;    // 4096x256
  float* t2      = ws + 3ull  * 1048576;    // 4096x256
  float* h_nodes = ws + 4ull  * 1048576;    // 4096x256
  float* gate    = ws + 5ull  * 1048576;    // 4096x1024
  float* graphl  = ws + 9ull  * 1048576;    // 4096x1024
  float* nodep   = ws + 13ull * 1048576;    // 4096x512
  float* tail    = ws + 2ull * MM;
  float* hgraph  = tail;                    // 1024: hh = [0:512), cc = [512:1024)
  float* seqemb  = tail + 1024;             // 512x64
  unsigned long long* amax = (unsigned long long*)(tail + 1024 + 32768);
  float* hh = hgraph;
  float* cc = hgraph + 512;

  const dim3 blk(256);

  // GCN / gating pipeline (all on the FP32 WMMA path). Grid = (N/64, M/128).
  wmma_gemm_f32<0><<<dim3(64, 32), blk, 0, stream>>>(d_tensor, a_tensor, nullptr, t0,      4096, 4096, 4096);
  wmma_gemm_f32<0><<<dim3(64, 32), blk, 0, stream>>>(t0,       d_tensor, nullptr, dmat,    4096, 4096, 4096);
  wmma_gemm_f32<1><<<dim3(4,  32), blk, 0, stream>>>(f_tensor, W_feat,   b_feat,  f_nodes, 4096, 256,  64);
  wmma_gemm_f32<0><<<dim3(4,  32), blk, 0, stream>>>(dmat,     f_nodes,  nullptr, t1,      4096, 256,  4096);
  wmma_gemm_f32<1><<<dim3(4,  32), blk, 0, stream>>>(t1,       W_gcn1,   nullptr, relu_h,  4096, 256,  256);
  wmma_gemm_f32<0><<<dim3(4,  32), blk, 0, stream>>>(dmat,     relu_h,   nullptr, t2,      4096, 256,  4096);
  wmma_gemm_f32<0><<<dim3(4,  32), blk, 0, stream>>>(t2,       W_gcn2,   nullptr, h_nodes, 4096, 256,  256);
  wmma_gemm_f32<2><<<dim3(16, 32), blk, 0, stream>>>(h_nodes,  W_gate,   b_gate,  gate,    4096, 1024, 256);
  wmma_gemm_f32<0><<<dim3(16, 32), blk, 0, stream>>>(h_nodes,  W_graph,  b_graph, graphl,  4096, 1024, 256);
  wmma_gemm_f32<0><<<dim3(8,  32), blk, 0, stream>>>(h_nodes,  W_ne,     b_ne,    nodep,   4096, 512,  256);

  gated_colsum_kernel<<<4, 256, 0, stream>>>(gate, graphl, hgraph);
  seq_embed_kernel<<<128, 256, 0, stream>>>(x_tensor, W_seq, b_seq, seqemb);

  // Sequential LSTM scan: 512 steps, 2 launches/step (graph-capture safe).
  float* scores  = (float*)d_out;                 // 512 x 4096
  float* out_idx = scores + 512ull * 4096ull;     // 512 (argmax as float)
  for (int t = 0; t < 512; ++t) {
    lstm_cell_kernel<<<1, 1024, 0, stream>>>(seqemb, W_ih, W_hh, b_ih, b_hh,
                                             hh, cc, amax, t);
    z_argmax_kernel<<<512, blk, 0, stream>>>(nodep, hh, w_ge, b_ge,
                                             scores, amax, t);
  }
  idx_finalize_kernel<<<2, 256, 0, stream>>>(amax, out_idx);
}